// HonestBiCameralCRSN_24902220382745
// MI455X (gfx1250) — compile-verified
//
#include <hip/hip_runtime.h>

// ---------------- problem constants ----------------
#define TPB     256
#define NWAVES  8
#define ROWS    16          // rows per block
#define SUB     16          // threads per row (TPB/ROWS)
#define D       128
#define F       256         // 2*D
#define STK     32
#define DEPTH   8
#define NSYN    64
#define NSEM    128

#define EPSC    1e-5f
#define MEMD    0.95f
#define TEMPC   10.0f
#define CTXS    3.0f

typedef __attribute__((ext_vector_type(2))) float v2f;
typedef __attribute__((ext_vector_type(8))) float v8f;

// fp32 WMMA: D(16x16,f32) = A(16x4,f32) x B(4x16,f32) + C
__device__ __forceinline__ v8f wmma4(v2f a, v2f b, v8f c) {
  return __builtin_amdgcn_wmma_f32_16x16x4_f32(false, a, false, b, (short)0, c,
                                               false, false);
}

// ---------------- per-row reductions (SUB partials via LDS) ----------------
__device__ __forceinline__ float rowsum(float v, float (*red)[SUB], int row, int sub) {
  red[row][sub] = v;
  __syncthreads();
  float s = 0.f;
#pragma unroll
  for (int j = 0; j < SUB; ++j) s += red[row][j];
  __syncthreads();
  return s;
}

__device__ __forceinline__ float rowmax(float v, float (*red)[SUB], int row, int sub) {
  red[row][sub] = v;
  __syncthreads();
  float s = -3.4e38f;
#pragma unroll
  for (int j = 0; j < SUB; ++j) s = fmaxf(s, red[row][j]);
  __syncthreads();
  return s;
}

// ---------------- generic 16xNC = 16xKD @ KDxNC WMMA GEMM ----------------
// X,Y in LDS; W in global (row-major [KD x NC], or transposed gather if bT:
// B[k][n] = W[n*ldw + k], used for z @ cb^T with cb row-major [NC x KD]).
__device__ __forceinline__ void gemm_lds(const float* X, int ldx,
                                         const float* W, int ldw,
                                         float* Y, int ldy,
                                         const float* bias,
                                         int NC, int KD,
                                         bool doRelu, bool bT) {
  const int wave = threadIdx.x >> 5;
  const int lane = threadIdx.x & 31;
  const int m16  = lane & 15;
  const int kh   = (lane >> 4) << 1;   // lanes 0-15: k+0/k+1, lanes 16-31: k+2/k+3
  const int nct  = NC >> 4;
  for (int t = wave; t < nct; t += NWAVES) {
    const int col0 = t << 4;
    v8f acc = {};
    const float* xr = X + m16 * ldx;
    for (int k0 = 0; k0 < KD; k0 += 4) {
      v2f a, b;
      a[0] = xr[k0 + kh];
      a[1] = xr[k0 + kh + 1];
      if (bT) {
        const float* wr = W + (size_t)(col0 + m16) * ldw;
        b[0] = wr[k0 + kh];
        b[1] = wr[k0 + kh + 1];
      } else {
        b[0] = W[(size_t)(k0 + kh) * ldw + col0 + m16];
        b[1] = W[(size_t)(k0 + kh + 1) * ldw + col0 + m16];
      }
      acc = wmma4(a, b, acc);
    }
    // C/D layout: VGPR j -> row j (lanes 0-15) / row 8+j (lanes 16-31), col = lane&15
    const int rb = (lane >> 4) << 3;
    const int c  = col0 + m16;
    const float bv = bias ? bias[c] : 0.f;
#pragma unroll
    for (int j = 0; j < 8; ++j) {
      float v = acc[j] + bv;
      if (doRelu) v = fmaxf(v, 0.f);
      Y[(rb + j) * ldy + c] = v;
    }
  }
}

// complex matmul: (r+ii) @ (Wr + i Wi): r2 = r@Wr - i@Wi, i2 = i@Wr + r@Wi
// zf LDS [16][256] = [r|i]; out LDS [16][256] = [r2|i2]. one 16x16 col tile / wave.
__device__ __forceinline__ void gemm_complex(const float* zfp, const float* Wr,
                                             const float* Wi, float* out) {
  const int wave = threadIdx.x >> 5;
  const int lane = threadIdx.x & 31;
  const int m16  = lane & 15;
  const int kh   = (lane >> 4) << 1;
  const int col0 = wave << 4;
  v8f cr = {}, ci = {};
  const float* xr = zfp + m16 * F;
  for (int k0 = 0; k0 < D; k0 += 4) {
    v2f ar, ai, ain, bwr, bwi;
    ar[0]  = xr[k0 + kh];         ar[1]  = xr[k0 + kh + 1];
    ai[0]  = xr[D + k0 + kh];     ai[1]  = xr[D + k0 + kh + 1];
    ain[0] = -ai[0];              ain[1] = -ai[1];    // lowers to v_xor sign-flip
    bwr[0] = Wr[(k0 + kh) * D + col0 + m16];
    bwr[1] = Wr[(k0 + kh + 1) * D + col0 + m16];
    bwi[0] = Wi[(k0 + kh) * D + col0 + m16];
    bwi[1] = Wi[(k0 + kh + 1) * D + col0 + m16];
    cr = wmma4(ar,  bwr, cr);
    cr = wmma4(ain, bwi, cr);
    ci = wmma4(ai,  bwr, ci);
    ci = wmma4(ar,  bwi, ci);
  }
  const int rb = (lane >> 4) << 3;
#pragma unroll
  for (int j = 0; j < 8; ++j) {
    out[(rb + j) * F + col0 + m16]     = cr[j];
    out[(rb + j) * F + D + col0 + m16] = ci[j];
  }
}

// ---------------- one VQ stage (ctx-MLP + distances + argmax) ----------------
__device__ __forceinline__ void vq_stage(
    float* zfp, float* bA, float* bB,
    float (*red)[SUB], int (*redi)[SUB],
    const float* w1, const float* b1, const float* g, const float* be,
    const float* w2, const float* b2, const float* wp, const float* bp,
    const float* cb, const float* adj, const float* cbn,
    int NC, int* idxarr, int row, int sub, float znorm, float sg_gate) {
  const float invF = 1.0f / (float)F;
  // h1 = zf3 @ w1 + b1 -> bufA
  gemm_lds(zfp, F, w1, F, bA, F, b1, F, F, false, false);
  __syncthreads();
  // layernorm * g + be, relu (in place)
  float ps = 0.f;
  for (int q = sub; q < F; q += SUB) ps += bA[row * F + q];
  const float hm = rowsum(ps, red, row, sub) * invF;
  float pv = 0.f;
  for (int q = sub; q < F; q += SUB) { float dd = bA[row * F + q] - hm; pv += dd * dd; }
  const float hi = rsqrtf(rowsum(pv, red, row, sub) * invF + 1e-5f);
  for (int q = sub; q < F; q += SUB) {
    float h = (bA[row * F + q] - hm) * hi * g[q] + be[q];
    bA[row * F + q] = fmaxf(h, 0.f);
  }
  __syncthreads();
  // h2 = relu(h1 @ w2 + b2) -> bufB   (w2: [256 x 128])
  gemm_lds(bA, F, w2, NSEM, bB, NSEM, b2, NSEM, F, true, false);
  __syncthreads();
  // cvals = h2 @ wp + bp -> bufA      (wp: [128 x NC])
  gemm_lds(bB, NSEM, wp, NC, bA, F, bp, NC, D, false, false);
  __syncthreads();
  // dot = zf3 @ cb^T -> bufB          (cb: [NC x 256], transposed gather)
  gemm_lds(zfp, F, cb, F, bB, NSEM, nullptr, NC, F, false, true);
  __syncthreads();
  const float invNC = 1.0f / (float)NC;
  for (int j = sub; j < NC; j += SUB) {
    float dval = znorm + cbn[j] - 2.0f * bB[row * NSEM + j];
    dval = fminf(fmaxf(dval, 0.0f), 10000.0f);
    bB[row * NSEM + j] = -dval;          // raw logits
  }
  __syncthreads();
  // LN of logits
  float ls = 0.f;
  for (int j = sub; j < NC; j += SUB) ls += bB[row * NSEM + j];
  const float lm = rowsum(ls, red, row, sub) * invNC;
  float lv = 0.f;
  for (int j = sub; j < NC; j += SUB) { float dd = bB[row * NSEM + j] - lm; lv += dd * dd; }
  const float li = rsqrtf(rowsum(lv, red, row, sub) * invNC + 1e-5f);
  // LN of ctx values
  float cs = 0.f;
  for (int j = sub; j < NC; j += SUB) cs += bA[row * F + j];
  const float cm = rowsum(cs, red, row, sub) * invNC;
  float cv = 0.f;
  for (int j = sub; j < NC; j += SUB) { float dd = bA[row * F + j] - cm; cv += dd * dd; }
  const float ci = rsqrtf(rowsum(cv, red, row, sub) * invNC + 1e-5f);
  // LN of softmax(adj[prev_idx]) (mean of softmax == 1/NC)
  const float* arow = adj + (size_t)idxarr[row] * NC;
  float am = -3.4e38f;
  for (int j = sub; j < NC; j += SUB) am = fmaxf(am, arow[j]);
  const float amax = rowmax(am, red, row, sub);
  float es = 0.f;
  for (int j = sub; j < NC; j += SUB) es += __expf(arow[j] - amax);
  const float isum = 1.0f / rowsum(es, red, row, sub);
  float av = 0.f;
  for (int j = sub; j < NC; j += SUB) {
    float dd = __expf(arow[j] - amax) * isum - invNC;
    av += dd * dd;
  }
  const float ai = rsqrtf(rowsum(av, red, row, sub) * invNC + 1e-5f);
  // combine + argmax (first-index tie-break, like jnp.argmax)
  float best = -3.4e38f; int bidx = 0x7fffffff;
  for (int j = sub; j < NC; j += SUB) {
    float lnl = (bB[row * NSEM + j] - lm) * li;
    float lnc = (bA[row * F + j] - cm) * ci;
    float lna = (__expf(arow[j] - amax) * isum - invNC) * ai;
    float sc  = lnl + lna * sg_gate + CTXS * lnc;
    if (sc > best || (sc == best && j < bidx)) { best = sc; bidx = j; }
  }
  red[row][sub] = best; redi[row][sub] = bidx;
  __syncthreads();
  if (sub == 0) {
    float bb = red[row][0]; int bi = redi[row][0];
#pragma unroll
    for (int j = 1; j < SUB; ++j)
      if (red[row][j] > bb || (red[row][j] == bb && redi[row][j] < bi)) {
        bb = red[row][j]; bi = redi[row][j];
      }
    idxarr[row] = bi;
  }
  __syncthreads();
}

// ---------------- fused recurrent kernel: one block = 16 rows, 8 depth steps ----
__global__ __launch_bounds__(TPB) void crsn_kernel(
    const float* __restrict__ zr, const float* __restrict__ zi,
    const float* __restrict__ memory0, const float* __restrict__ ptr0,
    const float* __restrict__ Wr, const float* __restrict__ Wi,
    const float* __restrict__ ln_scale, const float* __restrict__ ln_shift,
    const float* __restrict__ Wc, const float* __restrict__ bc,
    const float* __restrict__ cb_syn, const float* __restrict__ cb_sem,
    const float* __restrict__ adj_syn, const float* __restrict__ adj_sem,
    const float* __restrict__ graph_gate,
    const float* __restrict__ Wh, const float* __restrict__ bh,
    const int* __restrict__ prev_idx_syn, const int* __restrict__ prev_idx_sem,
    const float* __restrict__ cs_w1, const float* __restrict__ cs_b1,
    const float* __restrict__ cs_g,  const float* __restrict__ cs_be,
    const float* __restrict__ cs_w2, const float* __restrict__ cs_b2,
    const float* __restrict__ cs_wp, const float* __restrict__ cs_bp,
    const float* __restrict__ cm_w1, const float* __restrict__ cm_b1,
    const float* __restrict__ cm_g,  const float* __restrict__ cm_be,
    const float* __restrict__ cm_w2, const float* __restrict__ cm_b2,
    const float* __restrict__ cm_wp, const float* __restrict__ cm_bp,
    float* __restrict__ out, float* __restrict__ wsmem) {
  __shared__ float zf[ROWS][F];        // carry (r|i) -> zf -> zf3 -> zq
  __shared__ float bufA[ROWS][F];      // r2|i2 ; h1 ; cvals
  __shared__ float bufB[ROWS][NSEM];   // mag ; h2 ; logits
  __shared__ float ptrs[ROWS][STK];
  __shared__ float nptr[ROWS][STK];
  __shared__ float ctrl[ROWS][4];
  __shared__ float red[ROWS][SUB];
  __shared__ int   redi[ROWS][SUB];
  __shared__ int   idxs[ROWS];
  __shared__ int   idxm[ROWS];
  __shared__ float cbn_syn[NSYN];
  __shared__ float cbn_sem[NSEM];

  const int tid  = threadIdx.x;
  const int row  = tid >> 4;           // 0..15
  const int sub  = tid & (SUB - 1);    // 0..15
  const int grow = blockIdx.x * ROWS + row;

  float* zfp = &zf[0][0];
  float* bAp = &bufA[0][0];
  float* bBp = &bufB[0][0];

  // ---- init carry state ----
  for (int q = sub; q < D; q += SUB) {
    zf[row][q]     = zr[(size_t)grow * D + q];
    zf[row][D + q] = zi[(size_t)grow * D + q];
  }
  for (int q = sub; q < STK; q += SUB) ptrs[row][q] = ptr0[(size_t)grow * STK + q];
  if (sub == 0) {
    idxs[row] = prev_idx_syn[grow];
    idxm[row] = prev_idx_sem[grow];
  }
  // codebook squared norms (constant across depth)
  for (int t = tid; t < NSYN + NSEM; t += TPB) {
    const float* c = (t < NSYN) ? (cb_syn + (size_t)t * F)
                                : (cb_sem + (size_t)(t - NSYN) * F);
    float s = 0.f;
    for (int k = 0; k < F; ++k) s += c[k] * c[k];
    if (t < NSYN) cbn_syn[t] = s; else cbn_sem[t - NSYN] = s;
  }
  __syncthreads();

  const float sg = 1.0f / (1.0f + __expf(-graph_gate[0]));
  const float invD = 1.0f / (float)D;

  // Halting accumulator + remainder live entirely in registers: each thread
  // owns features q = sub + j*SUB of its row; p is broadcast to the whole row
  // by the rowsum reduction, so every thread tracks an identical rem copy.
  float acc[16];
#pragma unroll
  for (int j = 0; j < 16; ++j) acc[j] = 0.f;
  float rem = 1.0f;

  for (int dep = 0; dep < DEPTH; ++dep) {
    // ---- 1. complex matmul: zf=(r|i) -> bufA=(r2|i2) ----
    gemm_complex(zfp, Wr, Wi, bAp);
    __syncthreads();

    // ---- 2. magnitude layernorm + phase reconstruction -> zf ----
    float psum = 0.f;
    for (int q = sub; q < D; q += SUB) {
      float rr = bufA[row][q], ii = bufA[row][D + q];
      float mg = sqrtf(rr * rr + ii * ii) + EPSC;
      bufB[row][q] = mg;
      psum += mg;
    }
    const float mgm = rowsum(psum, red, row, sub) * invD;
    float pvar = 0.f;
    for (int q = sub; q < D; q += SUB) { float dd = bufB[row][q] - mgm; pvar += dd * dd; }
    const float mvar = fmaxf(rowsum(pvar, red, row, sub) * invD, 1e-5f);
    const float minv = rsqrtf(mvar + EPSC);
    for (int q = sub; q < D; q += SUB) {
      float rr = bufA[row][q], ii = bufA[row][D + q];
      float nm = (bufB[row][q] - mgm) * minv * ln_scale[q] + ln_shift[q];
      float ph = atan2f(ii, rr + 1e-9f);
      zf[row][q]     = nm * cosf(ph);
      zf[row][D + q] = nm * sinf(ph);
    }
    __syncthreads();

    // ---- 3. controller: softmax(zf @ Wc + bc) + stack pointer math ----
    if (tid < ROWS) {
      const int r = tid;
      float l0 = bc[0], l1 = bc[1], l2 = bc[2];
      for (int k = 0; k < F; ++k) {
        float z = zf[r][k];
        l0 += z * Wc[k * 3 + 0];
        l1 += z * Wc[k * 3 + 1];
        l2 += z * Wc[k * 3 + 2];
      }
      float mx = fmaxf(l0, fmaxf(l1, l2));
      float e0 = __expf(l0 - mx), e1 = __expf(l1 - mx), e2 = __expf(l2 - mx);
      float is = 1.f / (e0 + e1 + e2);
      ctrl[r][0] = e0 * is; ctrl[r][1] = e1 * is; ctrl[r][2] = e2 * is;
      // normalize pointer, shift, sharpen-softmax
      float s = EPSC;
      for (int q = 0; q < STK; ++q) s += ptrs[r][q];
      float invs = 1.f / s;
      for (int q = 0; q < STK; ++q) ptrs[r][q] *= invs;
      float lg[STK], lmx = -3.4e38f;
      const float push = ctrl[r][0], pop = ctrl[r][1], noop = ctrl[r][2];
      for (int q = 0; q < STK; ++q) {
        float up = (q == 0) ? 0.f : ptrs[r][q - 1];
        float dn = (q == STK - 1) ? 0.f : ptrs[r][q + 1];
        float l = (push * up + pop * dn + noop * ptrs[r][q]) * TEMPC;
        lg[q] = l; lmx = fmaxf(lmx, l);
      }
      float se = 0.f;
      for (int q = 0; q < STK; ++q) { lg[q] = __expf(lg[q] - lmx); se += lg[q]; }
      float ise = 1.f / se;
      for (int q = 0; q < STK; ++q) nptr[r][q] = lg[q] * ise;
    }
    __syncthreads();

    // ---- 4. stack memory stream (HBM-bound part): 16 feats/thread, float4 ----
    {
      const float* msrc = (dep == 0) ? memory0 : wsmem;
      const int f0 = sub * 16;
      const float push = ctrl[row][0];
      float racc[16];
#pragma unroll
      for (int q = 0; q < 16; ++q) racc[q] = 0.f;
      const float4* zv = reinterpret_cast<const float4*>(&zf[row][f0]);
      for (int s = 0; s < STK; ++s) {
        const float wm = push * ((s == 0) ? 0.f : ptrs[row][s - 1]);
        const float np = nptr[row][s];
        const float dk = (1.f - wm) * MEMD;
        const size_t base = ((size_t)grow * STK + (size_t)s) * F + f0;
        const float4* mi = reinterpret_cast<const float4*>(msrc + base);
        float4* mo = reinterpret_cast<float4*>(wsmem + base);
        if (s + 1 < STK) __builtin_prefetch(msrc + base + F, 0, 0);
#pragma unroll
        for (int q = 0; q < 4; ++q) {
          float4 m = mi[q];
          float4 z = zv[q];
          m.x = fminf(fmaxf(wm * z.x + m.x * dk, -1000.f), 1000.f);
          m.y = fminf(fmaxf(wm * z.y + m.y * dk, -1000.f), 1000.f);
          m.z = fminf(fmaxf(wm * z.z + m.z * dk, -1000.f), 1000.f);
          m.w = fminf(fmaxf(wm * z.w + m.w * dk, -1000.f), 1000.f);
          mo[q] = m;
          racc[4 * q + 0] += m.x * np;
          racc[4 * q + 1] += m.y * np;
          racc[4 * q + 2] += m.z * np;
          racc[4 * q + 3] += m.w * np;
        }
      }
      __syncthreads();
      // r3 = r2 + read ; carry new pointer
#pragma unroll
      for (int q = 0; q < 16; ++q) zf[row][f0 + q] += racc[q];
      for (int q = sub; q < STK; q += SUB) ptrs[row][q] = nptr[row][q];
      __syncthreads();
    }

    // ---- 5. |z|^2 per row, then the two VQ stages ----
    float zp = 0.f;
    for (int q = sub; q < F; q += SUB) { float z = zf[row][q]; zp += z * z; }
    const float znorm = rowsum(zp, red, row, sub);

    vq_stage(zfp, bAp, bBp, red, redi,
             cs_w1, cs_b1, cs_g, cs_be, cs_w2, cs_b2, cs_wp, cs_bp,
             cb_syn, adj_syn, cbn_syn, NSYN, idxs, row, sub, znorm, sg);
    vq_stage(zfp, bAp, bBp, red, redi,
             cm_w1, cm_b1, cm_g, cm_be, cm_w2, cm_b2, cm_wp, cm_bp,
             cb_sem, adj_sem, cbn_sem, NSEM, idxm, row, sub, znorm, sg);

    // ---- 6. zq = 0.5*(cb_syn[is] + cb_sem[im]); halting accumulate (regs) ----
    {
      const float* cbs = cb_syn + (size_t)idxs[row] * F;
      const float* cbm = cb_sem + (size_t)idxm[row] * F;
      float hdot = 0.f;
#pragma unroll
      for (int j = 0; j < 16; ++j) {
        const int q = sub + j * SUB;
        float zq = 0.5f * (cbs[q] + cbm[q]);
        zf[row][q] = zq;                // new carry
        hdot += zq * Wh[q];
      }
      const float hs = rowsum(hdot, red, row, sub);
      const float p = 1.0f / (1.0f + __expf(-(hs + bh[0])));
#pragma unroll
      for (int j = 0; j < 16; ++j) {
        const int q = sub + j * SUB;
        acc[j] += rem * p * zf[row][q];
      }
      rem *= (1.0f - p);
      __syncthreads();
    }
  }

  // ---- final single store: out = acc + rem * zf ----
  {
    float* orow = out + (size_t)grow * F;
#pragma unroll
    for (int j = 0; j < 16; ++j) {
      const int q = sub + j * SUB;
      orow[q] = acc[j] + rem * zf[row][q];
    }
  }
}

// ---------------- host launcher ----------------
extern "C" void kernel_launch(void* const* d_in, const int* in_sizes, int n_in,
                              void* d_out, int out_size, void* d_ws, size_t ws_size,
                              hipStream_t stream) {
  const float* zr        = (const float*)d_in[0];
  const float* zi        = (const float*)d_in[1];
  const float* memory0   = (const float*)d_in[2];
  const float* ptr0      = (const float*)d_in[3];
  const float* Wr        = (const float*)d_in[4];
  const float* Wi        = (const float*)d_in[5];
  const float* ln_scale  = (const float*)d_in[6];
  const float* ln_shift  = (const float*)d_in[7];
  const float* Wc        = (const float*)d_in[8];
  const float* bc        = (const float*)d_in[9];
  const float* cb_syn    = (const float*)d_in[10];
  const float* cb_sem    = (const float*)d_in[11];
  const float* adj_syn   = (const float*)d_in[12];
  const float* adj_sem   = (const float*)d_in[13];
  const float* ggate     = (const float*)d_in[14];
  const float* Wh        = (const float*)d_in[15];
  const float* bh        = (const float*)d_in[16];
  const int*   pidx_s    = (const int*)d_in[17];
  const int*   pidx_m    = (const int*)d_in[18];
  const float* cs_w1     = (const float*)d_in[19];
  const float* cs_b1     = (const float*)d_in[20];
  const float* cs_g      = (const float*)d_in[21];
  const float* cs_be     = (const float*)d_in[22];
  const float* cs_w2     = (const float*)d_in[23];
  const float* cs_b2     = (const float*)d_in[24];
  const float* cs_wp     = (const float*)d_in[25];
  const float* cs_bp     = (const float*)d_in[26];
  const float* cm_w1     = (const float*)d_in[27];
  const float* cm_b1     = (const float*)d_in[28];
  const float* cm_g      = (const float*)d_in[29];
  const float* cm_be     = (const float*)d_in[30];
  const float* cm_w2     = (const float*)d_in[31];
  const float* cm_b2     = (const float*)d_in[32];
  const float* cm_wp     = (const float*)d_in[33];
  const float* cm_bp     = (const float*)d_in[34];

  const int N = in_sizes[0] / D;        // 8192
  dim3 grid(N / ROWS), block(TPB);
  crsn_kernel<<<grid, block, 0, stream>>>(
      zr, zi, memory0, ptr0, Wr, Wi, ln_scale, ln_shift, Wc, bc,
      cb_syn, cb_sem, adj_syn, adj_sem, ggate, Wh, bh, pidx_s, pidx_m,
      cs_w1, cs_b1, cs_g, cs_be, cs_w2, cs_b2, cs_wp, cs_bp,
      cm_w1, cm_b1, cm_g, cm_be, cm_w2, cm_b2, cm_wp, cm_bp,
      (float*)d_out, (float*)d_ws);
}